// OccupancyGrid_27745488733000
// MI455X (gfx1250) — compile-verified
//
#include <hip/hip_runtime.h>
#include <stdint.h>

// Occupancy grid pipeline for MI455X (gfx1250, wave32).
// Grid: 500 x 500 x 50 voxels. Z-dim (50) is bit-packed into one uint64_t per
// (x,y) column -> whole grid is 2 MB and lives in the 192 MB L2.
#define GX 500
#define GY 500
#define GZ 50
#define NCOL (GX * GY)

// Mirror of reference _voxel_indices: float voxel coords via IEEE divide,
// in-bounds iff 0 <= v <= dim-1 (float compare), then truncate toward zero.
__device__ __forceinline__ bool voxel_index(float px, float py, float pz,
                                            float mnx, float mny, float mnz, float vs,
                                            int& ix, int& iy, int& iz)
{
    float fx = (px - mnx) / vs;
    float fy = (py - mny) / vs;
    float fz = (pz - mnz) / vs;
    bool ok = (fx >= 0.0f) && (fx <= (float)(GX - 1))
           && (fy >= 0.0f) && (fy <= (float)(GY - 1))
           && (fz >= 0.0f) && (fz <= (float)(GZ - 1));
    ix = (int)fx;  // truncation == astype(int32) for the in-bounds (>=0) case
    iy = (int)fy;
    iz = (int)fz;
    return ok;
}

__global__ void og_zero_grid(uint64_t* __restrict__ g, int n)
{
    int i = blockIdx.x * blockDim.x + threadIdx.x;
    if (i < n) g[i] = 0ull;
}

__device__ __forceinline__ void og_mark(float px, float py, float pz,
                                        float mnx, float mny, float mnz, float vs,
                                        uint64_t* __restrict__ grid)
{
    int ix, iy, iz;
    if (voxel_index(px, py, pz, mnx, mny, mnz, vs, ix, iy, iz)) {
        uint64_t* col = &grid[ix * GY + iy];
        uint64_t  bit = 1ull << iz;
        // Idempotent OR: cheap L2 probe avoids redundant RMWs on hot voxels.
        if ((*col & bit) == 0ull) {
            atomicOr((unsigned long long*)col, (unsigned long long)bit);
        }
    }
}

// Stage 1: scatter. 4 points per thread -> three coalesced b128 loads.
__global__ void og_scatter4(const float4* __restrict__ p4, int npts,
                            const float* __restrict__ aabb,
                            const float* __restrict__ vsp,
                            uint64_t* __restrict__ grid)
{
    int t    = blockIdx.x * blockDim.x + threadIdx.x;
    int base = t * 4;
    if (base >= npts) return;

    // Stream-prefetch ~96 KB ahead (gfx1250 global_prefetch_b8).
    if (base + 4 * 2048 < npts) __builtin_prefetch(&p4[(size_t)(t + 2048) * 3], 0, 0);

    const float mnx = aabb[0], mny = aabb[1], mnz = aabb[2];
    const float vs  = vsp[0];

    if (base + 4 <= npts) {
        float4 a = p4[(size_t)t * 3 + 0];
        float4 b = p4[(size_t)t * 3 + 1];
        float4 c = p4[(size_t)t * 3 + 2];
        og_mark(a.x, a.y, a.z, mnx, mny, mnz, vs, grid);
        og_mark(a.w, b.x, b.y, mnx, mny, mnz, vs, grid);
        og_mark(b.z, b.w, c.x, mnx, mny, mnz, vs, grid);
        og_mark(c.y, c.z, c.w, mnx, mny, mnz, vs, grid);
    } else {
        const float* p = (const float*)p4;
        for (int k = base; k < npts; ++k) {
            size_t o = (size_t)k * 3;
            og_mark(p[o + 0], p[o + 1], p[o + 2], mnx, mny, mnz, vs, grid);
        }
    }
}

// Stage 2: 3x3x3 box dilation. XY part = OR of up to 9 neighbor columns
// (zero padding at borders); Z part = c | c<<1 | c>>1 on the packed column.
// Shift distributes over OR, so this is exactly conv3d(ones(3,3,3)) > 0.
__global__ void og_dilate(const uint64_t* __restrict__ g, uint64_t* __restrict__ d)
{
    int i = blockIdx.x * blockDim.x + threadIdx.x;
    if (i >= NCOL) return;
    int x = i / GY;
    int y = i - x * GY;

    uint64_t acc = 0ull;
#pragma unroll
    for (int dx = -1; dx <= 1; ++dx) {
        int xx = x + dx;
        if (xx < 0 || xx >= GX) continue;
        const uint64_t* row = &g[xx * GY];
#pragma unroll
        for (int dy = -1; dy <= 1; ++dy) {
            int yy = y + dy;
            if (yy < 0 || yy >= GY) continue;
            acc |= row[yy];
        }
    }
    // Raw grid only ever sets bits 0..49, so >>1 shifts in zeros from bit 50.
    d[i] = acc | (acc << 1) | (acc >> 1);
}

__device__ __forceinline__ float og_probe(float px, float py, float pz,
                                          float mnx, float mny, float mnz, float vs,
                                          const uint64_t* __restrict__ grid)
{
    int ix, iy, iz;
    bool ok  = voxel_index(px, py, pz, mnx, mny, mnz, vs, ix, iy, iz);
    bool occ = false;
    if (ok) occ = ((grid[ix * GY + iy] >> iz) & 1ull) != 0ull;
    return occ ? 1.0f : 0.0f;
}

// Stage 3: query gather. 4 queries per thread -> b128 loads + one b128 store.
__global__ void og_query4(const float4* __restrict__ q4, int nq,
                          const float* __restrict__ aabb,
                          const float* __restrict__ vsp,
                          const uint64_t* __restrict__ grid,
                          float* __restrict__ out)
{
    int t    = blockIdx.x * blockDim.x + threadIdx.x;
    int base = t * 4;
    if (base >= nq) return;

    if (base + 4 * 2048 < nq) __builtin_prefetch(&q4[(size_t)(t + 2048) * 3], 0, 0);

    const float mnx = aabb[0], mny = aabb[1], mnz = aabb[2];
    const float vs  = vsp[0];

    if (base + 4 <= nq) {
        float4 a = q4[(size_t)t * 3 + 0];
        float4 b = q4[(size_t)t * 3 + 1];
        float4 c = q4[(size_t)t * 3 + 2];
        float4 r;
        r.x = og_probe(a.x, a.y, a.z, mnx, mny, mnz, vs, grid);
        r.y = og_probe(a.w, b.x, b.y, mnx, mny, mnz, vs, grid);
        r.z = og_probe(b.z, b.w, c.x, mnx, mny, mnz, vs, grid);
        r.w = og_probe(c.y, c.z, c.w, mnx, mny, mnz, vs, grid);
        ((float4*)out)[t] = r;
    } else {
        const float* q = (const float*)q4;
        for (int k = base; k < nq; ++k) {
            size_t o = (size_t)k * 3;
            out[k] = og_probe(q[o + 0], q[o + 1], q[o + 2], mnx, mny, mnz, vs, grid);
        }
    }
}

extern "C" void kernel_launch(void* const* d_in, const int* in_sizes, int n_in,
                              void* d_out, int out_size, void* d_ws, size_t ws_size,
                              hipStream_t stream)
{
    const float4* pts  = (const float4*)d_in[0];  // [Np,3] f32 (16B-aligned base)
    const float4* qpts = (const float4*)d_in[1];  // [Nq,3] f32
    const float*  aabb = (const float*)d_in[2];   // [2,3] f32 (row 0 = min)
    const float*  vs   = (const float*)d_in[3];   // scalar f32

    const int np = in_sizes[0] / 3;
    const int nq = in_sizes[1] / 3;

    // Workspace layout: raw grid (2,000,000 B) then dilated grid, 256-B aligned.
    const size_t grid_bytes = (size_t)NCOL * sizeof(uint64_t);
    const size_t dil_off    = (grid_bytes + 255) & ~(size_t)255;
    uint64_t* raw = (uint64_t*)d_ws;
    uint64_t* dil = (uint64_t*)((char*)d_ws + dil_off);

    const int B = 256;  // 8 wave32 waves per block

    // 1) zero the raw grid (ws is poisoned / left over from prior call)
    og_zero_grid<<<(NCOL + B - 1) / B, B, 0, stream>>>(raw, NCOL);

    // 2) scatter points (64-bit global atomics into L2-resident grid)
    const int nt_s = (np + 3) / 4;
    og_scatter4<<<(nt_s + B - 1) / B, B, 0, stream>>>(pts, np, aabb, vs, raw);

    // 3) 3x3x3 dilation on bit-packed columns
    og_dilate<<<(NCOL + B - 1) / B, B, 0, stream>>>(raw, dil);

    // 4) query gather -> float 0/1 output
    const int nt_q = (nq + 3) / 4;
    og_query4<<<(nt_q + B - 1) / B, B, 0, stream>>>(qpts, nq, aabb, vs, dil, (float*)d_out);
}